// M1_72164040508114
// MI455X (gfx1250) — compile-verified
//
#include <hip/hip_runtime.h>
#include <math.h>

typedef float v2f __attribute__((ext_vector_type(2)));
typedef float v8f __attribute__((ext_vector_type(8)));

#define GIN_N 100000
#define GIN_E 1600000

// ---------------------------------------------------------------------------
// hpre = (1 + eps_gin[layer]) * x        (fused self-term + accumulator init)
// ---------------------------------------------------------------------------
__global__ void gin_init_hpre(const float* __restrict__ x,
                              const float* __restrict__ eps_gin, int layer,
                              float* __restrict__ hpre, int total) {
  float e = 1.0f + eps_gin[layer];
  for (int i = blockIdx.x * blockDim.x + threadIdx.x; i < total;
       i += gridDim.x * blockDim.x)
    hpre[i] = e * x[i];
}

// ---------------------------------------------------------------------------
// scatter-add: hpre[dst] += x[src]   (16 threads / edge, float4 per thread)
// ---------------------------------------------------------------------------
__global__ void gin_scatter(const float* __restrict__ x,
                            const int* __restrict__ src,
                            const int* __restrict__ dst,
                            float* __restrict__ hpre, int E) {
  int tid = blockIdx.x * blockDim.x + threadIdx.x;
  int e = tid >> 4;
  if (e >= E) return;
  int j4 = (tid & 15) << 2;
  int s = src[e], d = dst[e];
  float4 v = *(const float4*)(x + (size_t)s * 64 + j4);
  float* p = hpre + (size_t)d * 64 + j4;
  atomicAdd(p + 0, v.x);
  atomicAdd(p + 1, v.y);
  atomicAdd(p + 2, v.z);
  atomicAdd(p + 3, v.w);
}

__global__ void gin_zero(float* __restrict__ p, int n) {
  int i = blockIdx.x * blockDim.x + threadIdx.x;
  if (i < n) p[i] = 0.0f;
}

// ---------------------------------------------------------------------------
// Out[N,outC] = A[N,64] @ W[64,outC] + bias, with fused per-column sum/sumsq
// (BatchNorm statistics).  fp32 WMMA 16x16x4, K=64 -> 16 steps x 4 col tiles.
// block = 256 threads (8 waves), 128 rows per block, 64 columns per block.y.
// W is staged into LDS K-pair-packed so each B fragment is one ds_load_b64.
// ---------------------------------------------------------------------------
__global__ __launch_bounds__(256) void gin_gemm64_wmma(
    const float* __restrict__ A, const float* __restrict__ W,
    const float* __restrict__ bias, float* __restrict__ Out,
    float* __restrict__ gsum, float* __restrict__ gsq, int N, int outC) {
  __shared__ float Wp[64 * 64];       // pair-packed: Wp[(k/2)*64+c] = {W[k][c],W[k+1][c]}
  __shared__ float Xs[8 * 16 * 68];   // per-wave 16x64 A tiles, padded stride 68
  __shared__ float ssum[64];
  __shared__ float ssq[64];

  const int tid = threadIdx.x;
  const int lane = tid & 31;
  const int wid = tid >> 5;
  const int colGrp = blockIdx.y * 64;
  const int rowBase = blockIdx.x * 128 + wid * 16;

  // stage W[0:64, colGrp:colGrp+64] into LDS, packing K-pairs into float2
#pragma unroll
  for (int it = 0; it < 8; ++it) {
    int idx = it * 256 + tid;            // 0..2047
    int p = idx >> 6, c = idx & 63;      // pair 0..31, column 0..63
    v2f w;
    w.x = W[(size_t)(2 * p) * outC + colGrp + c];
    w.y = W[(size_t)(2 * p + 1) * outC + colGrp + c];
    *(v2f*)(Wp + (p * 64 + c) * 2) = w;
  }
  // each wave stages its own 16x64 A tile (coalesced float4)
#pragma unroll
  for (int it = 0; it < 8; ++it) {
    int idx = it * 32 + lane;            // 0..255
    int r = idx >> 4, c4 = (idx & 15) << 2;
    int gr = rowBase + r;
    if (gr >= N) gr = N - 1;             // clamp; masked at store/stats
    float4 v = *(const float4*)(A + (size_t)gr * 64 + c4);
    *(float4*)(Xs + (wid * 16 + r) * 68 + c4) = v;
  }
  if (tid < 64) { ssum[tid] = 0.0f; ssq[tid] = 0.0f; }
  __syncthreads();

  const int m = lane & 15;               // row (A) / column (B,C) within tile
  const int khalf = lane >> 4;           // lanes 16-31 hold K+2/K+3
  const float* xrow = Xs + (wid * 16 + m) * 68;

  v8f acc[4] = {{}, {}, {}, {}};

#pragma unroll
  for (int kk = 0; kk < 16; ++kk) {
    v2f a = *(const v2f*)(xrow + kk * 4 + khalf * 2);        // ds_load_b64
    const int pb = (kk * 2 + khalf) * 128;                   // pair base (floats)
#pragma unroll
    for (int t = 0; t < 4; ++t) {
      v2f b = *(const v2f*)(Wp + pb + (t * 16 + m) * 2);     // ds_load_b64
      acc[t] = __builtin_amdgcn_wmma_f32_16x16x4_f32(
          false, a, false, b, (short)0, acc[t], false, false);
    }
  }

  // epilogue: bias, store, fused BN statistics
  const int rhi = khalf << 3;            // rows v..v+7 vs v+8..v+15
#pragma unroll
  for (int t = 0; t < 4; ++t) {
    int col = colGrp + t * 16 + m;
    float bcol = bias[col];
    float s = 0.0f, s2 = 0.0f;
#pragma unroll
    for (int v = 0; v < 8; ++v) {
      int row = rowBase + v + rhi;
      float val = acc[t][v] + bcol;
      if (row < N) {
        Out[(size_t)row * outC + col] = val;
        s += val;
        s2 += val * val;
      }
    }
    s += __shfl_xor(s, 16, 32);
    s2 += __shfl_xor(s2, 16, 32);
    if (lane < 16) {
      atomicAdd(&ssum[t * 16 + m], s);
      atomicAdd(&ssq[t * 16 + m], s2);
    }
  }
  __syncthreads();
  if (tid < 64) {
    atomicAdd(&gsum[colGrp + tid], ssum[tid]);
    atomicAdd(&gsq[colGrp + tid], ssq[tid]);
  }
}

// ---------------------------------------------------------------------------
// scale = gamma * rsqrt(var + eps); shift = beta - mean * scale
// ---------------------------------------------------------------------------
__global__ void gin_bnstats_finalize(const float* __restrict__ gsum,
                                     const float* __restrict__ gsq,
                                     const float* __restrict__ gamma,
                                     const float* __restrict__ beta,
                                     float* __restrict__ scale,
                                     float* __restrict__ shift, int C,
                                     float invN) {
  int c = threadIdx.x;
  if (c < C) {
    float mean = gsum[c] * invN;
    float var = gsq[c] * invN - mean * mean;   // biased, matches jnp.var
    float sc = gamma[c] * rsqrtf(var + 1e-5f);
    scale[c] = sc;
    shift[c] = beta[c] - mean * sc;
  }
}

// in-place BN + LeakyReLU(0.01), C == 64
__global__ void gin_bnlrelu64(float* __restrict__ h,
                              const float* __restrict__ scale,
                              const float* __restrict__ shift,
                              long long total) {
  for (long long i = (long long)blockIdx.x * blockDim.x + threadIdx.x;
       i < total; i += (long long)gridDim.x * blockDim.x) {
    int c = (int)(i & 63);
    float v = scale[c] * h[i] + shift[c];
    h[i] = v > 0.0f ? v : 0.01f * v;
  }
}

// ---------------------------------------------------------------------------
// head: out[n] = sum_j lrelu(scale[j]*hF[n,j]+shift[j]) * Wf2[j] + bf2
// one wave per node, 128-wide dot, shfl tree reduction
// ---------------------------------------------------------------------------
__global__ __launch_bounds__(256) void gin_head(
    const float* __restrict__ hF, const float* __restrict__ scale,
    const float* __restrict__ shift, const float* __restrict__ Wf2,
    const float* __restrict__ bf2, float* __restrict__ out, int N) {
  int lane = threadIdx.x & 31;
  int wid = threadIdx.x >> 5;
  int n = blockIdx.x * 8 + wid;
  if (n >= N) return;
  float acc = 0.0f;
#pragma unroll
  for (int q = 0; q < 4; ++q) {
    int j = lane + q * 32;
    float v = hF[(size_t)n * 128 + j];
    v = scale[j] * v + shift[j];
    v = v > 0.0f ? v : 0.01f * v;
    acc += v * Wf2[j];
  }
#pragma unroll
  for (int off = 16; off; off >>= 1) acc += __shfl_xor(acc, off, 32);
  if (lane == 0) out[n] = acc + bf2[0];
}

// ---------------------------------------------------------------------------
extern "C" void kernel_launch(void* const* d_in, const int* in_sizes, int n_in,
                              void* d_out, int out_size, void* d_ws,
                              size_t ws_size, hipStream_t stream) {
  (void)in_sizes; (void)n_in; (void)out_size; (void)ws_size;

  const float* x   = (const float*)d_in[0];
  const int* esrc  = (const int*)d_in[1];
  const int* edst  = (const int*)d_in[2];
  const float* W1  = (const float*)d_in[3];
  const float* b1  = (const float*)d_in[4];
  const float* g1  = (const float*)d_in[5];
  const float* be1 = (const float*)d_in[6];
  const float* W2  = (const float*)d_in[7];
  const float* b2  = (const float*)d_in[8];
  const float* eps = (const float*)d_in[9];
  const float* gbn = (const float*)d_in[10];
  const float* bbn = (const float*)d_in[11];
  const float* Wf1 = (const float*)d_in[12];
  const float* bf1 = (const float*)d_in[13];
  const float* gf  = (const float*)d_in[14];
  const float* bef = (const float*)d_in[15];
  const float* Wf2 = (const float*)d_in[16];
  const float* bf2 = (const float*)d_in[17];
  float* outp = (float*)d_out;
  float* ws = (float*)d_ws;

  const int N = GIN_N, E = GIN_E;
  const size_t nd = (size_t)N * 64;

  float* hpre = ws;             // [N,64]
  float* t1   = ws + nd;        // [N,64]
  float* xcur = ws + 2 * nd;    // [N,64]
  float* hF   = ws;             // [N,128] aliases hpre+t1 (dead at head stage)
  float* stats = ws + 3 * nd;   // sum[128] sq[128] scale[128] shift[128]
  float* gsum = stats;
  float* gsq  = stats + 128;
  float* scl  = stats + 256;
  float* shf  = stats + 384;

  dim3 blk(256);
  const int gElem = (int)((nd + 255) / 256);
  const int gScat = (int)(((size_t)E * 16 + 255) / 256);
  const int gRows = (N + 127) / 128;
  const float invN = 1.0f / (float)N;

  for (int i = 0; i < 3; ++i) {
    const float* xin = (i == 0) ? x : xcur;
    gin_init_hpre<<<gElem, blk, 0, stream>>>(xin, eps, i, hpre, (int)nd);
    gin_scatter<<<gScat, blk, 0, stream>>>(xin, esrc, edst, hpre, E);

    gin_zero<<<1, 256, 0, stream>>>(gsum, 256);
    gin_gemm64_wmma<<<dim3(gRows, 1), blk, 0, stream>>>(
        hpre, W1 + (size_t)i * 64 * 64, b1 + i * 64, t1, gsum, gsq, N, 64);
    gin_bnstats_finalize<<<1, 64, 0, stream>>>(gsum, gsq, g1 + i * 64,
                                               be1 + i * 64, scl, shf, 64, invN);
    gin_bnlrelu64<<<gElem, blk, 0, stream>>>(t1, scl, shf, (long long)nd);

    gin_zero<<<1, 256, 0, stream>>>(gsum, 256);
    gin_gemm64_wmma<<<dim3(gRows, 1), blk, 0, stream>>>(
        t1, W2 + (size_t)i * 64 * 64, b2 + i * 64, xcur, gsum, gsq, N, 64);
    if (i < 2) {
      gin_bnstats_finalize<<<1, 64, 0, stream>>>(
          gsum, gsq, gbn + i * 64, bbn + i * 64, scl, shf, 64, invN);
      gin_bnlrelu64<<<gElem, blk, 0, stream>>>(xcur, scl, shf, (long long)nd);
    }
  }

  // final head: [N,64] @ [64,128] + BN stats, then fused BN+lrelu+dot(Wf2)
  gin_zero<<<1, 256, 0, stream>>>(gsum, 256);
  gin_gemm64_wmma<<<dim3(gRows, 2), blk, 0, stream>>>(xcur, Wf1, bf1, hF, gsum,
                                                      gsq, N, 128);
  gin_bnstats_finalize<<<1, 128, 0, stream>>>(gsum, gsq, gf, bef, scl, shf, 128,
                                              invN);
  gin_head<<<(N + 7) / 8, blk, 0, stream>>>(hF, scl, shf, Wf2, bf2, outp, N);
}